// PGO_6305011991227
// MI455X (gfx1250) — compile-verified
//
#include <hip/hip_runtime.h>
#include <hip/hip_bf16.h>
#include <math.h>
#include <stdint.h>

#define EPS 1e-6f
#define BLOCK 256

// gfx1250 async global->LDS copy, tracked on ASYNCcnt.
//   global_load_async_to_lds_b128 vdst(lds byte offset), v[addr:addr+1](global), off
// Inline asm bypasses the clang builtin's HIP address-space typing entirely
// (per bridge doc guidance) and is portable across both toolchains.
__device__ __forceinline__ void async_b128_to_lds(const void* gptr, void* lds_ptr) {
    unsigned int lds_off = (unsigned int)(uintptr_t)lds_ptr;  // low 32 bits = LDS offset
    unsigned long long gaddr = (unsigned long long)(uintptr_t)gptr;
    asm volatile("global_load_async_to_lds_b128 %0, %1, off"
                 :: "v"(lds_off), "v"(gaddr)
                 : "memory");
}

__device__ __forceinline__ void wait_asynccnt0() {
    asm volatile("s_wait_asynccnt 0" ::: "memory");
}

struct V3 { float x, y, z; };
struct V4 { float x, y, z, w; };
struct Sim3 { V3 t; V4 q; float s; };

__device__ __forceinline__ V3 v3add(V3 a, V3 b) { return V3{a.x + b.x, a.y + b.y, a.z + b.z}; }
__device__ __forceinline__ V3 v3scale(V3 a, float s) { return V3{a.x * s, a.y * s, a.z * s}; }
__device__ __forceinline__ V3 v3cross(V3 a, V3 b) {
    return V3{a.y * b.z - a.z * b.y, a.z * b.x - a.x * b.z, a.x * b.y - a.y * b.x};
}
__device__ __forceinline__ float v3dot(V3 a, V3 b) { return a.x * b.x + a.y * b.y + a.z * b.z; }

__device__ __forceinline__ V3 quat_rotate(V4 q, V3 v) {
    V3 qv{q.x, q.y, q.z};
    V3 uv  = v3cross(qv, v);
    V3 uuv = v3cross(qv, uv);
    return v3add(v, v3scale(v3add(v3scale(uv, q.w), uuv), 2.0f));
}

__device__ __forceinline__ V4 quat_mul(V4 a, V4 b) {
    return V4{a.w * b.x + a.x * b.w + a.y * b.z - a.z * b.y,
              a.w * b.y - a.x * b.z + a.y * b.w + a.z * b.x,
              a.w * b.z + a.x * b.y - a.y * b.x + a.z * b.w,
              a.w * b.w - a.x * b.x - a.y * b.y - a.z * b.z};
}

__device__ __forceinline__ Sim3 sim3_inv(Sim3 T) {
    V4 qi{-T.q.x, -T.q.y, -T.q.z, T.q.w};
    float si = 1.0f / T.s;
    V3 ti = v3scale(quat_rotate(qi, T.t), -si);
    return Sim3{ti, qi, si};
}

__device__ __forceinline__ Sim3 sim3_mul(Sim3 A, Sim3 B) {
    return Sim3{v3add(A.t, v3scale(quat_rotate(A.q, B.t), A.s)),
                quat_mul(A.q, B.q), A.s * B.s};
}

// r = [tau(3), phi(3), sigma]; W tau = t solved via Cramer (3x3, well-conditioned).
__device__ __forceinline__ void sim3_log(Sim3 T, float r[7]) {
    V3 qv{T.q.x, T.q.y, T.q.z};
    float qw = T.q.w;
    float nv = sqrtf(v3dot(qv, qv));
    float theta_r = 2.0f * atan2f(nv, qw);
    bool small_r = nv < EPS;
    float fac = small_r ? 2.0f : (theta_r / nv);
    V3 phi = v3scale(qv, fac);

    float sigma = logf(T.s);
    float th2a  = v3dot(phi, phi);   // actual theta^2 (used for Phi@Phi)
    float theta = sqrtf(th2a);

    bool sig_small = fabsf(sigma) < EPS;
    bool th_small  = theta < EPS;
    float sg = sig_small ? 1.0f : sigma;
    float th = th_small  ? 1.0f : theta;
    float scl = expf(sigma);
    float th2 = th * th, sg2 = sg * sg;

    float C = sig_small ? 1.0f : ((scl - 1.0f) / sg);
    float sth = sinf(th), cth = cosf(th);
    float A_ss = th_small ? 0.5f : ((1.0f - cth) / th2);
    float B_ss = th_small ? (1.0f / 6.0f) : ((th - sth) / (th2 * th));
    float a = scl * sth, b = scl * cth, c = th2 + sg2;
    float A_g = (a * sg + (1.0f - b) * th) / (th * c);
    float B_g = (C - ((b - 1.0f) * sg + a * th) / c) / th2;
    float A_ts = ((sg - 1.0f) * scl + 1.0f) / sg2;
    float B_ts = (scl * (sg2 - 2.0f * sg + 2.0f) - 2.0f) / (2.0f * sg2 * sg);
    float A = sig_small ? A_ss : (th_small ? A_ts : A_g);
    float B = sig_small ? B_ss : (th_small ? B_ts : B_g);

    // W = C*I + A*Phi + B*(phi phi^T - th2a*I)
    float Cb = C - B * th2a;
    float x = phi.x, y = phi.y, z = phi.z;
    float m00 = Cb + B * x * x, m01 = -A * z + B * x * y, m02 =  A * y + B * x * z;
    float m10 =  A * z + B * x * y, m11 = Cb + B * y * y, m12 = -A * x + B * y * z;
    float m20 = -A * y + B * x * z, m21 =  A * x + B * y * z, m22 = Cb + B * z * z;

    float t0 = T.t.x, t1 = T.t.y, t2 = T.t.z;
    float c00 = m11 * m22 - m12 * m21;
    float c01 = m10 * m22 - m12 * m20;
    float c02 = m10 * m21 - m11 * m20;
    float det = m00 * c00 - m01 * c01 + m02 * c02;
    float idet = 1.0f / det;
    float u1 = t1 * m22 - m12 * t2;
    float u2 = m10 * t2 - t1 * m20;
    float u3 = m11 * t2 - t1 * m21;
    float d0 = t0 * c00 - m01 * u1 + m02 * (t1 * m21 - m11 * t2);
    float d1 = m00 * u1 - t0 * c01 + m02 * u2;
    float d2 = m00 * u3 - m01 * u2 + t0 * c02;

    r[0] = d0 * idet; r[1] = d1 * idet; r[2] = d2 * idet;
    r[3] = phi.x; r[4] = phi.y; r[5] = phi.z; r[6] = sigma;
}

// Sim3 row layout: [tx,ty,tz, qx,qy,qz,qw, s] = 2 x float4 (rows are 32B-aligned).
__device__ __forceinline__ Sim3 load_sim3(const float4* p) {
    float4 a = p[0];
    float4 b = p[1];
    return Sim3{V3{a.x, a.y, a.z}, V4{a.w, b.x, b.y, b.z}, b.w};
}

__global__ __launch_bounds__(BLOCK) void pgo_residual_kernel(
    const float4* __restrict__ Twc,     // N_FRAME x 2 float4
    const float4* __restrict__ Tpri,    // (N-1) x 2 float4
    const float4* __restrict__ Todo,    // (N-1) x 2 float4
    const float*  __restrict__ pw,      // (N-1) x 7
    const float*  __restrict__ ow,      // (N-1) x 7
    const int*    __restrict__ edges,   // (N-1) x 2
    const float4* __restrict__ Tlc,     // (N-1) x 2 float4
    float* __restrict__ out,            // (N-1) x 7
    int n)
{
    __shared__ float4 sh[BLOCK * 6];    // 24 KB: per-lane staging of Tpri/Todo/Tlc rows

    int i  = blockIdx.x * BLOCK + threadIdx.x;
    int ii = (i < n) ? i : (n - 1);     // clamp so every lane does safe loads

    // Gather indices first; prefetch the L2-resident random rows (Twc = 32 MB << 192 MB L2).
    int e0 = edges[2 * (size_t)ii + 0];
    int e1 = edges[2 * (size_t)ii + 1];
    __builtin_prefetch((const void*)(Twc + 2 * (size_t)e0), 0, 1);
    __builtin_prefetch((const void*)(Twc + 2 * (size_t)e1), 0, 1);

    {   // Stream the three per-row Sim3 operands into LDS on ASYNCcnt (async b128 x6).
        float4* dst = &sh[(size_t)threadIdx.x * 6];
        const float4* s0 = Tpri + 2 * (size_t)ii;
        const float4* s1 = Todo + 2 * (size_t)ii;
        const float4* s2 = Tlc  + 2 * (size_t)ii;
        async_b128_to_lds(s0 + 0, dst + 0);
        async_b128_to_lds(s0 + 1, dst + 1);
        async_b128_to_lds(s1 + 0, dst + 2);
        async_b128_to_lds(s1 + 1, dst + 3);
        async_b128_to_lds(s2 + 0, dst + 4);
        async_b128_to_lds(s2 + 1, dst + 5);
    }

    // Overlap: sequential-chain and gathered loads + compose math while async copies fly.
    Sim3 Ti = load_sim3(Twc + 2 * (size_t)ii);
    Sim3 Tj = load_sim3(Twc + 2 * (size_t)ii + 2);
    Sim3 Ta = load_sim3(Twc + 2 * (size_t)e0);
    Sim3 Tb = load_sim3(Twc + 2 * (size_t)e1);

    Sim3 delta = sim3_mul(sim3_inv(Ti), Tj);
    Sim3 dlc   = sim3_mul(sim3_inv(Ta), Tb);

    // Async copies done -> read staged operands from LDS (ds_load_b128).
    wait_asynccnt0();
    const float4* stg = &sh[(size_t)threadIdx.x * 6];
    Sim3 Tp = load_sim3(stg + 0);
    Sim3 To = load_sim3(stg + 2);
    Sim3 Tl = load_sim3(stg + 4);

    float rp[7], ro[7], rl[7];
    sim3_log(sim3_mul(delta, Tp), rp);
    sim3_log(sim3_mul(delta, To), ro);
    sim3_log(sim3_mul(dlc,   Tl), rl);

    if (i < n) {
        const float* pwr = pw + 7 * (size_t)i;
        const float* owr = ow + 7 * (size_t)i;
        float* o = out + 7 * (size_t)i;
#pragma unroll
        for (int k = 0; k < 7; ++k)
            o[k] = rp[k] * pwr[k] + ro[k] * owr[k] + rl[k];
    }
}

extern "C" void kernel_launch(void* const* d_in, const int* in_sizes, int n_in,
                              void* d_out, int out_size, void* d_ws, size_t ws_size,
                              hipStream_t stream) {
    const float4* Twc   = (const float4*)d_in[0];
    const float4* Tpri  = (const float4*)d_in[1];
    const float4* Todo  = (const float4*)d_in[2];
    const float*  pw    = (const float*)d_in[3];
    const float*  ow    = (const float*)d_in[4];
    const int*    edges = (const int*)d_in[5];
    const float4* Tlc   = (const float4*)d_in[6];
    float* out = (float*)d_out;

    int n = in_sizes[5] / 2;  // N_LC = N_FRAME - 1
    int grid = (n + BLOCK - 1) / BLOCK;
    pgo_residual_kernel<<<grid, BLOCK, 0, stream>>>(Twc, Tpri, Todo, pw, ow, edges, Tlc, out, n);
}